// WrapSimpleRetriSysClassifierThreshold_19971597927204
// MI455X (gfx1250) — compile-verified
//
#include <hip/hip_runtime.h>
#include <stdint.h>

#define NB_B 16384
#define NB_K 1000
#define NB_N 1000000
#define NB_C 100

// ---------------------------------------------------------------------------
// Kernel 1: compact int32 labels -> uint8 so the 16.4M random gathers read a
// 1MB table that stays resident in L2/WGP$ (4x less gather traffic, 4x more
// labels per cacheline).
// ---------------------------------------------------------------------------
__global__ void __launch_bounds__(256)
compact_labels_kernel(const int* __restrict__ in, unsigned char* __restrict__ out, int n) {
  int i = blockIdx.x * blockDim.x + threadIdx.x;
  if (i < n) out[i] = (unsigned char)in[i];
}

// CDNA5 async global->LDS copy (ASYNCcnt path), inline asm per CDNA5_HIP.md
// recommendation (portable across ROCm7.2 / amdgpu-toolchain builtin arities).
// Each lane copies 16 bytes: from the global address in gaddr to the LDS address in ldsoff.
#define ASYNC_B128(ldsoff, gaddr)                                            \
  asm volatile("global_load_async_to_lds_b128 %0, %1, off"                   \
               :: "v"(ldsoff), "v"(gaddr) : "memory")

#define WAIT_ASYNC(n) asm volatile("s_wait_asynccnt " #n ::: "memory")

// ---------------------------------------------------------------------------
// Kernel 2: one wave32 per row. Double-buffered async staging of the index
// stream into LDS, byte-gather of labels (L2-resident), per-wave LDS
// histogram via ds_add_u32, wave reduction, normalized write-out.
// ---------------------------------------------------------------------------
template <typename LT>
__global__ void __launch_bounds__(256)
hist_probs_kernel(const int* __restrict__ tidx, const LT* __restrict__ labels,
                  float* __restrict__ out) {
  __shared__ unsigned hist[8][128];                    // per-wave histogram (padded)
  __shared__ __align__(16) int idxbuf[8][2][128];      // per-wave double buffer

  const int lane = threadIdx.x & 31;
  const int wave = threadIdx.x >> 5;
  const int row  = blockIdx.x * 8 + wave;

  // Zero this wave's histogram (wave-local; same-wave DS ops are in-order,
  // so no barrier is needed before the atomics).
  #pragma unroll
  for (int c = lane; c < 128; c += 32) hist[wave][c] = 0u;

  const int* rowp = tidx + (size_t)row * NB_K;

  // LDS byte offsets of this lane's 16B slot in each buffer.
  const unsigned lds0 = (unsigned)(size_t)(&idxbuf[wave][0][0]) + (unsigned)lane * 16u;
  const unsigned lds1 = (unsigned)(size_t)(&idxbuf[wave][1][0]) + (unsigned)lane * 16u;

  // Prime: async-stage chunk 0 (128 int32 indices = 512B per wave) into buf0.
  {
    unsigned long long g = (unsigned long long)(rowp + lane * 4);
    ASYNC_B128(lds0, g);
  }

  constexpr int NFULL = NB_K / 128;  // 7 full chunks (896 indices)
  #pragma unroll 1
  for (int ch = 0; ch < NFULL; ++ch) {
    if (ch + 1 < NFULL) {
      // Prefetch next chunk into the other buffer, then wait until only it
      // is outstanding (async loads complete in order => current buf ready).
      unsigned long long g = (unsigned long long)(rowp + (ch + 1) * 128 + lane * 4);
      ASYNC_B128(((ch + 1) & 1) ? lds1 : lds0, g);
      WAIT_ASYNC(1);
    } else {
      WAIT_ASYNC(0);
    }
    const int4 v = *(const int4*)(&idxbuf[wave][ch & 1][lane * 4]);  // ds_load_b128
    unsigned c0 = (unsigned)labels[v.x];
    unsigned c1 = (unsigned)labels[v.y];
    unsigned c2 = (unsigned)labels[v.z];
    unsigned c3 = (unsigned)labels[v.w];
    atomicAdd(&hist[wave][c0], 1u);   // ds_add_u32 (no return)
    atomicAdd(&hist[wave][c1], 1u);
    atomicAdd(&hist[wave][c2], 1u);
    atomicAdd(&hist[wave][c3], 1u);
  }

  // Tail: remaining 104 indices via plain coalesced loads.
  for (int k = NFULL * 128 + lane; k < NB_K; k += 32) {
    unsigned c = (unsigned)labels[rowp[k]];
    atomicAdd(&hist[wave][c], 1u);
  }

  // Row total (robustly computed; equals K here). Wave32 butterfly reduce.
  unsigned s = hist[wave][lane] + hist[wave][lane + 32] + hist[wave][lane + 64];
  if (lane < NB_C - 96) s += hist[wave][lane + 96];
  #pragma unroll
  for (int off = 16; off > 0; off >>= 1) s += __shfl_xor(s, off, 32);
  const float inv = 1.0f / (float)(s > 0u ? s : 1u);

  float* orow = out + (size_t)row * NB_C;
  orow[lane]      = (float)hist[wave][lane]      * inv;
  orow[lane + 32] = (float)hist[wave][lane + 32] * inv;
  orow[lane + 64] = (float)hist[wave][lane + 64] * inv;
  if (lane < NB_C - 96) orow[lane + 96] = (float)hist[wave][lane + 96] * inv;
}

extern "C" void kernel_launch(void* const* d_in, const int* in_sizes, int n_in,
                              void* d_out, int out_size, void* d_ws, size_t ws_size,
                              hipStream_t stream) {
  const int* tidx   = (const int*)d_in[0];   // threshold_index [B*K]
  const int* labels = (const int*)d_in[1];   // database_labels [N]
  float* out        = (float*)d_out;         // probs [B*C]

  const int rows   = in_sizes[0] / NB_K;     // 16384
  const int nlab   = in_sizes[1];            // 1000000
  const int blocks = rows / 8;               // 8 waves (rows) per 256-thread block

  if (d_ws != nullptr && ws_size >= (size_t)nlab) {
    unsigned char* lab8 = (unsigned char*)d_ws;
    compact_labels_kernel<<<(nlab + 255) / 256, 256, 0, stream>>>(labels, lab8, nlab);
    hist_probs_kernel<unsigned char><<<blocks, 256, 0, stream>>>(tidx, lab8, out);
  } else {
    hist_probs_kernel<int><<<blocks, 256, 0, stream>>>(tidx, labels, out);
  }
}